// EncoderAttention_6253472383006
// MI455X (gfx1250) — compile-verified
//
#include <hip/hip_runtime.h>

typedef __attribute__((ext_vector_type(2))) float v2f;
typedef __attribute__((ext_vector_type(8))) float v8f;
typedef __attribute__((ext_vector_type(4))) int   v4i;

#define INPUT  4096
#define DK     1024
#define S_ROWS 8192

// ---------------- CDNA5 async (memory -> LDS) staging support --------------
#ifndef __has_builtin
#define __has_builtin(x) 0
#endif

#if defined(__HIP_DEVICE_COMPILE__) && __has_builtin(__builtin_amdgcn_global_load_async_to_lds_b128)
#define HAVE_ASYNC 1
#else
#define HAVE_ASYNC 0
#endif

#if HAVE_ASYNC
// global_load_async_to_lds_b128: per-lane 16B, memory -> LDS, ASYNCcnt tracked.
// Param 0: addrspace(1) v4i*  (global);  Param 1: addrspace(3) v4i*  (LDS).
#define STAGE16(gp, lp)                                                        \
    __builtin_amdgcn_global_load_async_to_lds_b128(                            \
        (__attribute__((address_space(1))) v4i*)(uintptr_t)(const void*)(gp),  \
        (__attribute__((address_space(3))) v4i*)(void*)(lp), 0, 0)
#if __has_builtin(__builtin_amdgcn_s_wait_asynccnt)
#define WAIT_ASYNC(n) __builtin_amdgcn_s_wait_asynccnt(n)
#else
#define WAIT_ASYNC(n) asm volatile("s_wait_asynccnt %0" ::"i"(n))
#endif
#else
// Fallback: load to VGPR then ds_store_b128 (compiler inserts waits)
#define STAGE16(gp, lp) (*(float4*)(lp) = *(const float4*)(gp))
#define WAIT_ASYNC(n)   ((void)0)
#endif

// ---------------------------------------------------------------------------
// Row-major GEMV: y[row] = W[row,:] . x + b[row]   (wave per row, shfl reduce)
// Used for: query = Wq @ target + bq  and  out = Wv @ u + bv
// ---------------------------------------------------------------------------
__global__ __launch_bounds__(256) void k_rowgemv(const float* __restrict__ W,
                                                 const float* __restrict__ x,
                                                 const float* __restrict__ b,
                                                 float* __restrict__ y,
                                                 int ncols) {
    const int wave = threadIdx.x >> 5;
    const int lane = threadIdx.x & 31;
    const int row  = blockIdx.x * 8 + wave;
    const float4* wr = (const float4*)(W + (size_t)row * ncols);
    const float4* xv = (const float4*)x;
    float s = 0.f;
    for (int i = lane; i < (ncols >> 2); i += 32) {
        float4 w = wr[i], v = xv[i];
        s += w.x * v.x + w.y * v.y + w.z * v.z + w.w * v.w;
    }
    #pragma unroll
    for (int off = 16; off; off >>= 1) s += __shfl_down(s, off, 32);
    if (lane == 0) y[row] = s + b[row];
}

// ---------------------------------------------------------------------------
// t[i] = (1/32) * sum_j query[j] * Wk[j, i]    (coalesced column GEMV)
// 1/sqrt(d_k)=1/32 folded here; bk dropped (softmax is shift-invariant).
// ---------------------------------------------------------------------------
__global__ __launch_bounds__(256) void k2_t(const float* __restrict__ Wk,
                                            const float* __restrict__ q,
                                            float* __restrict__ t) {
    __shared__ float qs[DK];
    for (int i = threadIdx.x; i < DK; i += 256) qs[i] = q[i];
    __syncthreads();
    const int col = blockIdx.x * 256 + threadIdx.x;
    float s = 0.f;
    #pragma unroll 4
    for (int j = 0; j < DK; ++j) s += qs[j] * Wk[(size_t)j * INPUT + col];
    t[col] = s * 0.03125f;
}

// ---------------------------------------------------------------------------
// Partial logits via V_WMMA_F32_16X16X4_F32 with async-LDS tile staging.
// Wave = 16 encoder rows x 1024-K split. Tiles of 16x32 f32 are staged
// memory->LDS with GLOBAL_LOAD_ASYNC_TO_LDS_B128 (fully coalesced 128B rows),
// double-buffered per wave (no block barriers: each wave owns its region).
// LDS row stride 36 floats -> the 64 DWORDs of each ds_load_b64 fragment
// read hit 64 distinct banks (36*delta mod 64 is a multiple of 4).
// B = t[k..k+3] broadcast into all 16 columns; D[:,0] = row partials.
// ---------------------------------------------------------------------------
#define KT     32
#define ROWPAD 36
#define TILEF  (16 * ROWPAD)   // 576 floats per buffer

__global__ __launch_bounds__(256) void k3_logits_wmma(const float* __restrict__ enc,
                                                      const float* __restrict__ t,
                                                      float* __restrict__ part) {
    __shared__ float smem[8 * 2 * TILEF];   // 36 KB

    const int wave  = threadIdx.x >> 5;
    const int lane  = threadIdx.x & 31;
    const int gw    = blockIdx.x * 8 + wave;   // 0..2047
    const int tile  = gw >> 2;                 // 0..511  (16-row tiles)
    const int split = gw & 3;                  // 0..3    (K splits of 1024)
    const int r0    = tile * 16;
    const int k0    = split * 1024;
    const int m     = lane & 15;
    const int h     = lane >> 4;               // half-wave -> K pair select

    float* buf0 = &smem[(wave * 2 + 0) * TILEF];
    float* buf1 = &smem[(wave * 2 + 1) * TILEF];

    // staging lane map: 4 lane-groups of 8 -> 4 rows x full 128B per instr
    const int srow = lane >> 3;                // 0..3
    const int scol = lane & 7;                 // 0..7 (16B chunks)
    const float* gbase = enc + (size_t)(r0 + srow) * INPUT + k0 + scol * 4;
    const int    lbase = srow * ROWPAD + scol * 4;

    // consume pointers
    const float* bbase = t + k0 + 2 * h;

    v8f c = {};

    // prologue: stage tile 0 into buf0
    #pragma unroll
    for (int i = 0; i < 4; ++i)
        STAGE16(gbase + (size_t)(4 * i) * INPUT, buf0 + 4 * i * ROWPAD + lbase);

    for (int it = 0; it < 31; ++it) {
        float* cur = (it & 1) ? buf1 : buf0;
        float* nxt = (it & 1) ? buf0 : buf1;
        // prefetch next tile
        const float* gnext = gbase + (it + 1) * KT;
        #pragma unroll
        for (int i = 0; i < 4; ++i)
            STAGE16(gnext + (size_t)(4 * i) * INPUT, nxt + 4 * i * ROWPAD + lbase);
        WAIT_ASYNC(4);                          // current tile has landed
        const float* ap = cur + m * ROWPAD + 2 * h;
        const float* bp = bbase + it * KT;
        #pragma unroll
        for (int kl = 0; kl < KT; kl += 4) {
            v2f a = *(const v2f*)(ap + kl);
            v2f b = *(const v2f*)(bp + kl);
            c = __builtin_amdgcn_wmma_f32_16x16x4_f32(
                    false, a, false, b, (short)0, c, false, false);
        }
    }
    WAIT_ASYNC(0);                              // last tile (it=31, odd -> buf1)
    {
        const float* ap = buf1 + m * ROWPAD + 2 * h;
        const float* bp = bbase + 31 * KT;
        #pragma unroll
        for (int kl = 0; kl < KT; kl += 4) {
            v2f a = *(const v2f*)(ap + kl);
            v2f b = *(const v2f*)(bp + kl);
            c = __builtin_amdgcn_wmma_f32_16x16x4_f32(
                    false, a, false, b, (short)0, c, false, false);
        }
    }

    // D[m, n=0]: lane 0 holds M=0..7 in c[0..7]; lane 16 holds M=8..15.
    if (lane == 0) {
        #pragma unroll
        for (int g = 0; g < 8; ++g) part[split * S_ROWS + r0 + g] = c[g];
    } else if (lane == 16) {
        #pragma unroll
        for (int g = 0; g < 8; ++g) part[split * S_ROWS + r0 + 8 + g] = c[g];
    }
}

// ---------------------------------------------------------------------------
// Softmax over 8192 logits (sum of 4 deterministic K-split partials).
// ---------------------------------------------------------------------------
__global__ __launch_bounds__(1024) void k4_softmax(const float* __restrict__ part,
                                                   float* __restrict__ scores) {
    __shared__ float red[1024];
    const int tid = threadIdx.x;
    float l[8];
    float mx = -3.4e38f;
    #pragma unroll
    for (int j = 0; j < 8; ++j) {
        int r = tid + j * 1024;
        float v = part[r] + part[S_ROWS + r] + part[2 * S_ROWS + r] + part[3 * S_ROWS + r];
        l[j] = v;
        mx = fmaxf(mx, v);
    }
    red[tid] = mx; __syncthreads();
    for (int s = 512; s > 0; s >>= 1) {
        if (tid < s) red[tid] = fmaxf(red[tid], red[tid + s]);
        __syncthreads();
    }
    const float M = red[0]; __syncthreads();
    float sum = 0.f;
    #pragma unroll
    for (int j = 0; j < 8; ++j) { l[j] = __expf(l[j] - M); sum += l[j]; }
    red[tid] = sum; __syncthreads();
    for (int s = 512; s > 0; s >>= 1) {
        if (tid < s) red[tid] += red[tid + s];
        __syncthreads();
    }
    const float inv = 1.f / red[0];
    #pragma unroll
    for (int j = 0; j < 8; ++j) scores[tid + j * 1024] = l[j] * inv;
}

// ---------------------------------------------------------------------------
// u partials: upart[split, col] = sum_{r in split} scores[r] * enc[r, col]
// float4 per thread (global_load_b128), 32 row-splits x 4 col-blocks.
// ---------------------------------------------------------------------------
__global__ __launch_bounds__(256) void k5_upart(const float* __restrict__ enc,
                                                const float* __restrict__ scores,
                                                float* __restrict__ upart) {
    __shared__ float ss[256];
    const int cb    = blockIdx.x & 3;          // 4 col blocks of 1024
    const int split = blockIdx.x >> 2;         // 32 splits of 256 rows
    const int r0    = split * 256;
    if (threadIdx.x < 256) ss[threadIdx.x] = scores[r0 + threadIdx.x];
    __syncthreads();
    const int col = cb * 1024 + threadIdx.x * 4;
    const float* ep = enc + (size_t)r0 * INPUT + col;
    float4 s = {0.f, 0.f, 0.f, 0.f};
    #pragma unroll 8
    for (int r = 0; r < 256; ++r) {
        float4 e = *(const float4*)(ep + (size_t)r * INPUT);
        float w = ss[r];
        s.x += w * e.x; s.y += w * e.y; s.z += w * e.z; s.w += w * e.w;
    }
    *(float4*)(upart + split * INPUT + col) = s;
}

__global__ __launch_bounds__(256) void k5b_collapse(const float* __restrict__ upart,
                                                    float* __restrict__ u) {
    const int i = blockIdx.x * 256 + threadIdx.x;
    float s = 0.f;
    #pragma unroll
    for (int j = 0; j < 32; ++j) s += upart[j * INPUT + i];
    u[i] = s;
}

// ---------------------------------------------------------------------------
extern "C" void kernel_launch(void* const* d_in, const int* in_sizes, int n_in,
                              void* d_out, int out_size, void* d_ws, size_t ws_size,
                              hipStream_t stream) {
    const float* target = (const float*)d_in[0];
    const float* enc    = (const float*)d_in[1];
    const float* Wq     = (const float*)d_in[2];
    const float* bq     = (const float*)d_in[3];
    const float* Wk     = (const float*)d_in[4];
    // d_in[5] (bk) intentionally unused: softmax is invariant to a uniform shift.
    const float* Wv     = (const float*)d_in[6];
    const float* bv     = (const float*)d_in[7];
    float* out = (float*)d_out;

    float* ws     = (float*)d_ws;
    float* query  = ws;              // 1024
    float* t      = ws + 1024;       // 4096
    float* part   = ws + 5120;       // 4 * 8192 logit partials
    float* scores = ws + 37888;      // 8192
    float* upart  = ws + 46080;      // 32 * 4096 u partials
    float* u      = ws + 177152;     // 4096   (total ~708 KB)
    (void)in_sizes; (void)n_in; (void)out_size; (void)ws_size;

    k_rowgemv     <<<128, 256, 0, stream>>>(Wq, target, bq, query, INPUT);
    k2_t          <<< 16, 256, 0, stream>>>(Wk, query, t);
    k3_logits_wmma<<<256, 256, 0, stream>>>(enc, t, part);
    k4_softmax    <<<  1, 1024, 0, stream>>>(part, scores);
    k5_upart      <<<128, 256, 0, stream>>>(enc, scores, upart);
    k5b_collapse  <<< 16, 256, 0, stream>>>(upart, u);
    k_rowgemv     <<<128, 256, 0, stream>>>(Wv, u, bv, out, INPUT);
}